// BoxSeg_24043226923460
// MI455X (gfx1250) — compile-verified
//
#include <hip/hip_runtime.h>

#define M_PRED 100
#define N_GT   20
#define HW     640000                       // 800*800
#define MASK_BYTES (HW * 4)                 // 2,560,000 bytes per mask
#define TILE_BYTES 4096                     // 256 threads * 16B
#define TILES_PER_MASK (MASK_BYTES / TILE_BYTES)   // 625
#define TILES_PER_BLOCK 25
#define BLOCKS_PER_MASK (TILES_PER_MASK / TILES_PER_BLOCK) // 25
#define BATCH 5                             // async tiles per phase (25 = 5 batches)

// ---- CDNA5 async global<->LDS DMA (ASYNCcnt path, CDNA5 ISA) ----
__device__ __forceinline__ void async_load_b128(unsigned lds_addr, unsigned voff,
                                                unsigned long long saddr) {
    // GVS mode: copies 16 bytes from the global address saddr plus the 32-bit offset voff to the LDS address in lds_addr
    asm volatile("global_load_async_to_lds_b128 %0, %1, %2"
                 :: "v"(lds_addr), "v"(voff), "s"(saddr) : "memory");
}
__device__ __forceinline__ void async_store_b128(unsigned voff, unsigned lds_addr,
                                                 unsigned long long saddr) {
    asm volatile("global_store_async_from_lds_b128 %0, %1, %2"
                 :: "v"(voff), "v"(lds_addr), "s"(saddr) : "memory");
}
__device__ __forceinline__ void wait_asynccnt0() {
    asm volatile("s_wait_asynccnt 0x0" ::: "memory");
}
__device__ __forceinline__ void wait_dscnt0() {
    asm volatile("s_wait_dscnt 0x0" ::: "memory");
}

// ---------------- Kernel 1: IoU + faithful greedy match (tiny, 1 block) ----------------
__global__ void __launch_bounds__(128)
match_kernel(const float* __restrict__ pred_boxes,
             const float* __restrict__ pred_scores,
             const float* __restrict__ mask_score,
             const float* __restrict__ gt_boxes,
             float* __restrict__ out_tail,   // d_out + N*HW : [flags(N), scores(N)]
             int* __restrict__ gtm_out) {
    __shared__ float iou[M_PRED][N_GT];
    __shared__ float sc[M_PRED];
    __shared__ int   order[M_PRED];
    __shared__ float gx1[N_GT], gy1[N_GT], gx2[N_GT], gy2[N_GT], garea[N_GT];

    const int tid = threadIdx.x;
    if (tid < N_GT) {
        float x1 = gt_boxes[tid * 4 + 0], y1 = gt_boxes[tid * 4 + 1];
        float x2 = gt_boxes[tid * 4 + 2], y2 = gt_boxes[tid * 4 + 3];
        gx1[tid] = x1; gy1[tid] = y1; gx2[tid] = x2; gy2[tid] = y2;
        garea[tid] = (x2 - x1) * (y2 - y1);
    }
    if (tid < M_PRED) sc[tid] = pred_scores[tid];
    __syncthreads();

    if (tid < M_PRED) {
        float x1 = pred_boxes[tid * 4 + 0], y1 = pred_boxes[tid * 4 + 1];
        float x2 = pred_boxes[tid * 4 + 2], y2 = pred_boxes[tid * 4 + 3];
        float a1 = (x2 - x1) * (y2 - y1);
        for (int j = 0; j < N_GT; ++j) {
            float lx = fmaxf(x1, gx1[j]), ly = fmaxf(y1, gy1[j]);
            float rx = fminf(x2, gx2[j]), ry = fminf(y2, gy2[j]);
            float w = fmaxf(rx - lx, 0.0f), h = fmaxf(ry - ly, 0.0f);
            float inter = w * h;
            float uni = a1 + garea[j] - inter;
            iou[tid][j] = inter / fmaxf(uni, 1e-6f);
        }
        // stable descending argsort: rank = #{larger} + #{equal with smaller index}
        float si = sc[tid];
        int r = 0;
        for (int k = 0; k < M_PRED; ++k) {
            float sk = sc[k];
            r += (sk > si) || ((sk == si) && (k < tid));
        }
        order[r] = tid;
    }
    __syncthreads();

    if (tid == 0) {
        int gtm[N_GT];
        for (int j = 0; j < N_GT; ++j) gtm[j] = -1;
        for (int k = 0; k < M_PRED; ++k) {
            int i = order[k];
            float si = sc[i];
            float best = -__builtin_huge_valf();
            int m = 0;
            bool any = false;
            for (int j = 0; j < N_GT; ++j) {
                float v = iou[i][j];
                // faithful replica of the original `gtm[j] > 0` availability check
                bool valid = (gtm[j] <= 0) && (v >= 0.5f);
                any = any || valid;
                float mv = valid ? v : -__builtin_huge_valf();
                if (mv > best) { best = mv; m = j; }  // strict > => first max
            }
            if (any && (si >= 0.05f)) gtm[m] = i;
        }
        for (int j = 0; j < N_GT; ++j) {
            int g = gtm[j];
            gtm_out[j] = g;
            bool matched = (g > -1);
            int idx = matched ? g : 0;
            out_tail[j]        = matched ? mask_score[idx] : 0.0f;  // gt_masks_flags
            out_tail[N_GT + j] = matched ? sc[idx]         : 0.0f;  // out_pred_scores
        }
    }
}

// ------- Kernel 2: bandwidth-bound mask gather via async global->LDS->global DMA -------
// Batched double-buffered pipeline: up to 10 async b128 ops (5 KB) in flight per wave.
__global__ void __launch_bounds__(256)
gather_masks_kernel(const float* __restrict__ pred_masks,
                    const int* __restrict__ gtm,
                    float* __restrict__ out_masks) {
    const int tid  = threadIdx.x;
    const int mask = blockIdx.x / BLOCKS_PER_MASK;
    const int blk  = blockIdx.x % BLOCKS_PER_MASK;

    const int src = __builtin_amdgcn_readfirstlane(gtm[mask]);

    const unsigned long long chunk_off =
        (unsigned long long)blk * (TILES_PER_BLOCK * TILE_BYTES);
    unsigned long long dst = (unsigned long long)(out_masks) +
                             (unsigned long long)mask * MASK_BYTES + chunk_off;

    __shared__ __align__(16) unsigned char lbuf[2][BATCH][TILE_BYTES];  // 40 KB
    const unsigned voff = (unsigned)(tid * 16);
    const unsigned ldsG0 = (unsigned)(unsigned long long)(&lbuf[0][0][0]) + voff;
    const unsigned ldsG1 = (unsigned)(unsigned long long)(&lbuf[1][0][0]) + voff;

    if (src < 0) {
        // unmatched gt: zero-fill via async stores from one zeroed LDS tile.
        // Each lane only async-stores the 16B it wrote itself -> wave-local dep only.
        *(float4*)(&lbuf[0][0][tid * 16]) = make_float4(0.0f, 0.0f, 0.0f, 0.0f);
        wait_dscnt0();
#pragma unroll
        for (int t = 0; t < TILES_PER_BLOCK; ++t)
            async_store_b128(voff, ldsG0, dst + (unsigned long long)t * TILE_BYTES);
        return;  // stores drained by implicit S_WAIT_IDLE at S_ENDPGM
    }

    unsigned long long srcb = (unsigned long long)(pred_masks) +
                              (unsigned long long)src * MASK_BYTES + chunk_off;

    // prologue: batch 0 -> group 0
#pragma unroll
    for (int b = 0; b < BATCH; ++b)
        async_load_b128(ldsG0 + b * TILE_BYTES, voff,
                        srcb + (unsigned long long)b * TILE_BYTES);
    srcb += (unsigned long long)BATCH * TILE_BYTES;

#pragma unroll 1
    for (int bt = 0; bt < TILES_PER_BLOCK / BATCH; ++bt) {
        wait_asynccnt0();  // loads(bt) done; stores(bt-1) done -> safe to reuse groups
        const unsigned cur = (bt & 1) ? ldsG1 : ldsG0;
        const unsigned nxt = (bt & 1) ? ldsG0 : ldsG1;
#pragma unroll
        for (int b = 0; b < BATCH; ++b)
            async_store_b128(voff, cur + b * TILE_BYTES,
                             dst + (unsigned long long)b * TILE_BYTES);
        dst += (unsigned long long)BATCH * TILE_BYTES;
        if (bt + 1 < TILES_PER_BLOCK / BATCH) {
#pragma unroll
            for (int b = 0; b < BATCH; ++b)
                async_load_b128(nxt + b * TILE_BYTES, voff,
                                srcb + (unsigned long long)b * TILE_BYTES);
            srcb += (unsigned long long)BATCH * TILE_BYTES;
        }
    }
    // outstanding async stores drained by implicit S_WAIT_IDLE at S_ENDPGM
}

extern "C" void kernel_launch(void* const* d_in, const int* in_sizes, int n_in,
                              void* d_out, int out_size, void* d_ws, size_t ws_size,
                              hipStream_t stream) {
    (void)in_sizes; (void)n_in; (void)out_size; (void)ws_size;
    const float* pred_boxes  = (const float*)d_in[0];
    const float* pred_scores = (const float*)d_in[1];
    const float* mask_score  = (const float*)d_in[2];
    const float* pred_masks  = (const float*)d_in[3];
    const float* gt_boxes    = (const float*)d_in[4];
    float* out = (float*)d_out;
    int*   gtm = (int*)d_ws;

    match_kernel<<<1, 128, 0, stream>>>(pred_boxes, pred_scores, mask_score, gt_boxes,
                                        out + (size_t)N_GT * HW, gtm);
    gather_masks_kernel<<<N_GT * BLOCKS_PER_MASK, 256, 0, stream>>>(pred_masks, gtm, out);
}